// LossOT_63556926046295
// MI455X (gfx1250) — compile-verified
//
#include <hip/hip_runtime.h>
#include <hip/hip_bf16.h>

typedef __attribute__((ext_vector_type(2))) float v2f;
typedef __attribute__((ext_vector_type(8))) float v8f;

#define DD        1024      // H*W
#define CCH       3
#define NFR       5
#define NBATCH    8
#define S_PAIRS   24
#define NITER_SK  20
#define REG_F     0.5f
#define INVREG_F  2.0f      // 1/REG
#define NTHREADS  512
#define NWAVES    16        // 512 / 32 (wave32)
#define LOG_A     (-6.931471805599453f)   // -log(1024)

// Online log-sum-exp combine of (m1,s1) with a new term t (s-term weight 1)
__device__ __forceinline__ void lse_acc(float& m, float& s, float t) {
    float nm = fmaxf(m, t);
    s = s * __expf(m - nm) + __expf(t - nm);
    m = nm;
}
__device__ __forceinline__ void lse_merge(float& m, float& s, float om, float os) {
    float nm = fmaxf(m, om);
    s = s * __expf(m - nm) + os * __expf(om - nm);
    m = nm;
}

__global__ __launch_bounds__(NTHREADS)
void LossOT_sinkhorn_kernel(const float* __restrict__ burst,   // [B,N,C,D]
                            const float* __restrict__ gt,      // [B,C,D]
                            const int*   __restrict__ indices, // [S,4]
                            float*       __restrict__ lw)      // [S] partial loss*w
{
    __shared__ float s_ri[DD * 4];   // [d][c], c padded to 4 (K=4 for WMMA)
    __shared__ float s_rj[DD * 4];
    __shared__ float s_a2[DD];
    __shared__ float s_b2[DD];
    __shared__ float s_f[DD];
    __shared__ float s_g[DD];
    __shared__ float s_red[NTHREADS];
    __shared__ float s_wpart[NWAVES];
    __shared__ float s_w;

    const int s    = blockIdx.x;
    const int tid  = threadIdx.x;
    const int lane = tid & 31;
    const int wave = tid >> 5;

    const int bi = indices[s * 4 + 0];
    const int bj = indices[s * 4 + 1];
    const int fi = indices[s * 4 + 2];
    const int fj = indices[s * 4 + 3];

    // -------- Phase 0: build ri/rj (padded), squared norms, f=g=0, weight w --------
    float sum_r = 0.0f;
    for (int d = tid; d < DD; d += NTHREADS) {
        float r0 = gt[(bi * CCH + 0) * DD + d] - burst[(((bi * NFR + fi) * CCH) + 0) * DD + d];
        float r1 = gt[(bi * CCH + 1) * DD + d] - burst[(((bi * NFR + fi) * CCH) + 1) * DD + d];
        float r2 = gt[(bi * CCH + 2) * DD + d] - burst[(((bi * NFR + fi) * CCH) + 2) * DD + d];
        s_ri[d * 4 + 0] = r0; s_ri[d * 4 + 1] = r1; s_ri[d * 4 + 2] = r2; s_ri[d * 4 + 3] = 0.0f;
        s_a2[d] = r0 * r0 + r1 * r1 + r2 * r2;
        sum_r += r0 + r1 + r2;

        float q0 = gt[(bj * CCH + 0) * DD + d] - burst[(((bj * NFR + fj) * CCH) + 0) * DD + d];
        float q1 = gt[(bj * CCH + 1) * DD + d] - burst[(((bj * NFR + fj) * CCH) + 1) * DD + d];
        float q2 = gt[(bj * CCH + 2) * DD + d] - burst[(((bj * NFR + fj) * CCH) + 2) * DD + d];
        s_rj[d * 4 + 0] = q0; s_rj[d * 4 + 1] = q1; s_rj[d * 4 + 2] = q2; s_rj[d * 4 + 3] = 0.0f;
        s_b2[d] = q0 * q0 + q1 * q1 + q2 * q2;
        sum_r += q0 + q1 + q2;

        s_f[d] = 0.0f;
        s_g[d] = 0.0f;
    }
    s_red[tid] = sum_r;
    __syncthreads();
    for (int off = NTHREADS / 2; off > 0; off >>= 1) {
        if (tid < off) s_red[tid] += s_red[tid + off];
        __syncthreads();
    }
    if (tid == 0) s_w = s_red[0] / (2.0f * (float)(DD * CCH));  // 0.5*(mean_ri+mean_rj)
    __syncthreads();

    const int m    = lane & 15;
    const int k0   = (lane < 16) ? 0 : 2;   // K-half per ISA A/B fragment layout
    const int roff = (lane < 16) ? 0 : 8;   // C/D row-half per lane group

    // -------- Phase 1: Sinkhorn iterations (tiles of M via WMMA f32 16x16x4) --------
    for (int it = 0; it < NITER_SK; ++it) {
        // f[d] = reg*(loga - LSE_e((g[e] - M[d,e]) / reg))
        for (int rb = wave; rb < DD / 16; rb += NWAVES) {
            const int dbase = rb * 16;
            v2f A;
            A.x = s_ri[(dbase + m) * 4 + k0];
            A.y = s_ri[(dbase + m) * 4 + k0 + 1];
            float a2d[8];
            #pragma unroll
            for (int i = 0; i < 8; ++i) a2d[i] = s_a2[dbase + roff + i];
            float mx[8], sm[8];
            #pragma unroll
            for (int i = 0; i < 8; ++i) { mx[i] = -3.0e38f; sm[i] = 0.0f; }

            for (int eb = 0; eb < DD / 16; ++eb) {
                const int ebase = eb * 16;
                v2f Bf;
                Bf.x = s_rj[(ebase + m) * 4 + k0];
                Bf.y = s_rj[(ebase + m) * 4 + k0 + 1];
                v8f c = {};
                c = __builtin_amdgcn_wmma_f32_16x16x4_f32(false, A, false, Bf,
                                                          (short)0, c, false, false);
                const float ge  = s_g[ebase + m];
                const float b2e = s_b2[ebase + m];
                #pragma unroll
                for (int i = 0; i < 8; ++i) {
                    float Mde = a2d[i] + b2e - 2.0f * c[i];
                    lse_acc(mx[i], sm[i], (ge - Mde) * INVREG_F);
                }
            }
            // merge across the 16 lanes that share each row
            #pragma unroll
            for (int i = 0; i < 8; ++i) {
                #pragma unroll
                for (int msk = 1; msk < 16; msk <<= 1) {
                    float om = __shfl_xor(mx[i], msk, 32);
                    float os = __shfl_xor(sm[i], msk, 32);
                    lse_merge(mx[i], sm[i], om, os);
                }
            }
            if ((lane & 15) == 0) {
                #pragma unroll
                for (int i = 0; i < 8; ++i)
                    s_f[dbase + roff + i] = REG_F * (LOG_A - (mx[i] + __logf(sm[i])));
            }
        }
        __syncthreads();

        // g[e] = reg*(loga - LSE_d((f[d] - M[d,e]) / reg))
        for (int cb = wave; cb < DD / 16; cb += NWAVES) {
            const int ebase = cb * 16;
            v2f Bf;
            Bf.x = s_rj[(ebase + m) * 4 + k0];
            Bf.y = s_rj[(ebase + m) * 4 + k0 + 1];
            const float b2e = s_b2[ebase + m];
            float mx = -3.0e38f, sm = 0.0f;

            for (int db = 0; db < DD / 16; ++db) {
                const int dbase = db * 16;
                v2f A;
                A.x = s_ri[(dbase + m) * 4 + k0];
                A.y = s_ri[(dbase + m) * 4 + k0 + 1];
                v8f c = {};
                c = __builtin_amdgcn_wmma_f32_16x16x4_f32(false, A, false, Bf,
                                                          (short)0, c, false, false);
                #pragma unroll
                for (int i = 0; i < 8; ++i) {
                    const int d = dbase + roff + i;
                    float Mde = s_a2[d] + b2e - 2.0f * c[i];
                    lse_acc(mx, sm, (s_f[d] - Mde) * INVREG_F);
                }
            }
            { // merge lane L with L+16 (same column e)
                float om = __shfl_xor(mx, 16, 32);
                float os = __shfl_xor(sm, 16, 32);
                lse_merge(mx, sm, om, os);
            }
            if (lane < 16) s_g[ebase + lane] = REG_F * (LOG_A - (mx + __logf(sm)));
        }
        __syncthreads();
    }

    // -------- Phase 2: loss = sum_{d,e} exp((f+g-M)/reg) * M --------
    float lsum = 0.0f;
    for (int rb = wave; rb < DD / 16; rb += NWAVES) {
        const int dbase = rb * 16;
        v2f A;
        A.x = s_ri[(dbase + m) * 4 + k0];
        A.y = s_ri[(dbase + m) * 4 + k0 + 1];
        float a2d[8], fd[8];
        #pragma unroll
        for (int i = 0; i < 8; ++i) {
            a2d[i] = s_a2[dbase + roff + i];
            fd[i]  = s_f[dbase + roff + i];
        }
        for (int eb = 0; eb < DD / 16; ++eb) {
            const int ebase = eb * 16;
            v2f Bf;
            Bf.x = s_rj[(ebase + m) * 4 + k0];
            Bf.y = s_rj[(ebase + m) * 4 + k0 + 1];
            v8f c = {};
            c = __builtin_amdgcn_wmma_f32_16x16x4_f32(false, A, false, Bf,
                                                      (short)0, c, false, false);
            const float ge  = s_g[ebase + m];
            const float b2e = s_b2[ebase + m];
            #pragma unroll
            for (int i = 0; i < 8; ++i) {
                float Mde = a2d[i] + b2e - 2.0f * c[i];
                lsum += __expf((fd[i] + ge - Mde) * INVREG_F) * Mde;
            }
        }
    }
    #pragma unroll
    for (int msk = 1; msk < 32; msk <<= 1) lsum += __shfl_xor(lsum, msk, 32);
    if (lane == 0) s_wpart[wave] = lsum;
    __syncthreads();
    if (tid == 0) {
        float tot = 0.0f;
        #pragma unroll
        for (int w2 = 0; w2 < NWAVES; ++w2) tot += s_wpart[w2];
        lw[s] = tot * s_w;
    }
}

__global__ void LossOT_finalize_kernel(const float* __restrict__ lw,
                                       float* __restrict__ out)
{
    int t = threadIdx.x;
    float v = (t < S_PAIRS) ? lw[t] : 0.0f;
    #pragma unroll
    for (int msk = 16; msk >= 1; msk >>= 1) v += __shfl_xor(v, msk, 32);
    if (t == 0) out[0] = v / (float)S_PAIRS;
}

extern "C" void kernel_launch(void* const* d_in, const int* in_sizes, int n_in,
                              void* d_out, int out_size, void* d_ws, size_t ws_size,
                              hipStream_t stream) {
    const float* burst   = (const float*)d_in[0];  // [8,5,3,32,32]
    const float* gt_img  = (const float*)d_in[1];  // [8,3,32,32]
    const int*   indices = (const int*)d_in[2];    // [24,4]
    float* out = (float*)d_out;
    float* lw  = (float*)d_ws;                     // 24 floats of scratch

    LossOT_sinkhorn_kernel<<<S_PAIRS, NTHREADS, 0, stream>>>(burst, gt_img, indices, lw);
    LossOT_finalize_kernel<<<1, 32, 0, stream>>>(lw, out);
}